// GatherLayer_37082747633839
// MI455X (gfx1250) — compile-verified
//
#include <hip/hip_runtime.h>

// GatherLayer: out[b, j] = full_output[b, idx[b]*512 + j]
// B=16384, OUTPUT_SIZE=512, NB_ACTIONS=18. Pure bandwidth problem (~64 MB
// traffic -> ~2.8 us at 23.3 TB/s). CDNA5 path: async global<->LDS B128 copies.

#define OUTPUT_SIZE 512
#define NB_ACTIONS 18
#define ROW_STRIDE (OUTPUT_SIZE * NB_ACTIONS) /* 9216 floats */
#define ROWS_PER_BLOCK 2
#define CHUNKS_PER_ROW (OUTPUT_SIZE / 4) /* 128 x float4 per row */
#define THREADS (ROWS_PER_BLOCK * CHUNKS_PER_ROW) /* 256 = 8 wave32 */

// Builtin signature (from clang diagnostic): pointee is a 4 x i32 vector,
// global operand in addrspace(1), LDS operand in addrspace(3).
typedef int v4i __attribute__((ext_vector_type(4)));
typedef __attribute__((address_space(1))) v4i* gv4i_ptr; // global v4i*
typedef __attribute__((address_space(3))) v4i* lv4i_ptr; // LDS v4i*

#if __has_builtin(__builtin_amdgcn_global_load_async_to_lds_b128) && \
    __has_builtin(__builtin_amdgcn_global_store_async_from_lds_b128) && \
    __has_builtin(__builtin_amdgcn_s_wait_asynccnt)
#define HAVE_ASYNC_LDS 1
#else
#define HAVE_ASYNC_LDS 0
#endif

__global__ __launch_bounds__(THREADS) void gather_rows_kernel(
    const float* __restrict__ full, const int* __restrict__ idx,
    float* __restrict__ out, int batch) {
  const int tid = threadIdx.x;
  const int row_local = tid >> 7;  // 0..ROWS_PER_BLOCK-1, wave-uniform
  const int j4 = tid & (CHUNKS_PER_ROW - 1);  // float4 index within row
  const int b = blockIdx.x * ROWS_PER_BLOCK + row_local;
  if (b >= batch) return;

  const int a = idx[b];  // wave-uniform load (tid>>7 constant per wave32)
  const float* src =
      full + (size_t)b * ROW_STRIDE + (size_t)a * OUTPUT_SIZE + (size_t)j4 * 4;
  float* dst = out + (size_t)b * OUTPUT_SIZE + (size_t)j4 * 4;

#if HAVE_ASYNC_LDS
  __shared__ float smem[ROWS_PER_BLOCK * OUTPUT_SIZE];
  float* stage = &smem[row_local * OUTPUT_SIZE + j4 * 4];

  // 16 B/lane: global -> LDS via the async copy engine (ASYNCcnt-tracked).
  __builtin_amdgcn_global_load_async_to_lds_b128(
      (gv4i_ptr)(uintptr_t)src, (lv4i_ptr)(uintptr_t)stage,
      /*offset=*/0, /*cpol=*/0);
  // Each lane consumes exactly the LDS bytes its own load wrote; per-wave
  // asynccnt==0 is a sufficient ordering guarantee (no block barrier needed).
  __builtin_amdgcn_s_wait_asynccnt(0);
  // LDS -> global, also async (16 B/lane).
  __builtin_amdgcn_global_store_async_from_lds_b128(
      (gv4i_ptr)(uintptr_t)dst, (lv4i_ptr)(uintptr_t)stage,
      /*offset=*/0, /*cpol=*/0);
  __builtin_amdgcn_s_wait_asynccnt(0);
#else
  // Fallback: plain 16 B/lane register copy (global_load_b128/global_store_b128).
  const float4 v = *(const float4*)src;
  *(float4*)dst = v;
#endif
}

extern "C" void kernel_launch(void* const* d_in, const int* in_sizes, int n_in,
                              void* d_out, int out_size, void* d_ws,
                              size_t ws_size, hipStream_t stream) {
  const float* full = (const float*)d_in[0];
  const int* idx = (const int*)d_in[1];
  float* out = (float*)d_out;
  const int batch = in_sizes[1];  // one int index per batch row

  const int blocks = (batch + ROWS_PER_BLOCK - 1) / ROWS_PER_BLOCK;
  gather_rows_kernel<<<dim3(blocks), dim3(THREADS), 0, stream>>>(full, idx, out,
                                                                 batch);
}